// CTC_12610023981192
// MI455X (gfx1250) — compile-verified
//
#include <hip/hip_runtime.h>

#define T_DIM 2000
#define B_DIM 64
#define V_DIM 512
#define L_DIM 200
#define S_DIM 401        // 2*L + 1
#define NEGV  (-1e30f)

typedef __attribute__((ext_vector_type(2))) float v2f;
typedef __attribute__((ext_vector_type(4))) float v4f;
typedef __attribute__((ext_vector_type(8))) float v8f;

__device__ __forceinline__ float lae2(float a, float b) {
    float m = fmaxf(a, b);
    float d = fabsf(a - b);
    return m + __logf(1.0f + __expf(-d));
}

__device__ __forceinline__ float lae3(float a, float b, float c) {
    float m = fmaxf(fmaxf(a, b), c);
    float s = __expf(a - m) + __expf(b - m) + __expf(c - m);
    return m + __logf(s);
}

// ---------------------------------------------------------------------------
// Kernel 1: per-(t,b) row logsumexp.  One wave32 per 512-float row.
// HBM-bound streaming pass: non-temporal b128 loads + shuffle reductions.
// ---------------------------------------------------------------------------
__global__ __launch_bounds__(256) void ctc_lse_kernel(const float* __restrict__ x,
                                                      float* __restrict__ lse) {
    const int wave = threadIdx.x >> 5;
    const int lane = threadIdx.x & 31;
    const long long row = (long long)blockIdx.x * 8 + wave;   // row < T*B
    const v4f* rp = (const v4f*)(x + row * V_DIM);            // 128 float4 per row

    v4f d0 = __builtin_nontemporal_load(rp + lane);
    v4f d1 = __builtin_nontemporal_load(rp + lane + 32);
    v4f d2 = __builtin_nontemporal_load(rp + lane + 64);
    v4f d3 = __builtin_nontemporal_load(rp + lane + 96);

    float m = fmaxf(fmaxf(d0.x, d0.y), fmaxf(d0.z, d0.w));
    m = fmaxf(m, fmaxf(fmaxf(d1.x, d1.y), fmaxf(d1.z, d1.w)));
    m = fmaxf(m, fmaxf(fmaxf(d2.x, d2.y), fmaxf(d2.z, d2.w)));
    m = fmaxf(m, fmaxf(fmaxf(d3.x, d3.y), fmaxf(d3.z, d3.w)));
    #pragma unroll
    for (int off = 16; off >= 1; off >>= 1)
        m = fmaxf(m, __shfl_xor(m, off, 32));

    float s = __expf(d0.x - m) + __expf(d0.y - m) + __expf(d0.z - m) + __expf(d0.w - m)
            + __expf(d1.x - m) + __expf(d1.y - m) + __expf(d1.z - m) + __expf(d1.w - m)
            + __expf(d2.x - m) + __expf(d2.y - m) + __expf(d2.z - m) + __expf(d2.w - m)
            + __expf(d3.x - m) + __expf(d3.y - m) + __expf(d3.z - m) + __expf(d3.w - m);
    #pragma unroll
    for (int off = 16; off >= 1; off >>= 1)
        s += __shfl_xor(s, off, 32);

    if (lane == 0) lse[row] = m + __logf(s);
}

// ---------------------------------------------------------------------------
// Kernel 2: CTC forward scan.  One workgroup (416 thr = 13 waves) per batch.
// One lattice state per thread; alpha double-buffered in LDS -> exactly one
// barrier per time step.  Emission gathers are software-pipelined one step
// ahead so the global-load latency overlaps the barrier + lae3 work of the
// current step; global_prefetch_b8 runs at distance 2.
// ---------------------------------------------------------------------------
__global__ __launch_bounds__(416) void ctc_scan_kernel(const float* __restrict__ x,
                                                       const int* __restrict__ targets,
                                                       const float* __restrict__ lse,
                                                       float* __restrict__ nll) {
    __shared__ float alpha[2][S_DIM];

    const int b = blockIdx.x;
    const int s = threadIdx.x;
    const bool act = (s < S_DIM);

    // extended label + skip connection for this state (blank for even s)
    int  extIdx = 0;
    bool skip   = false;
    if (act && (s & 1)) {
        const int li = s >> 1;
        extIdx = targets[b * L_DIM + li];
        if (s >= 3) skip = (extIdx != targets[b * L_DIM + li - 1]);
    }

    // t = 0 initialization
    {
        const float* row0 = x + (long long)b * V_DIM;
        const float  l0   = lse[b];
        if (act) {
            float a0 = NEGV;
            if (s == 0)      a0 = row0[0] - l0;
            else if (s == 1) a0 = row0[extIdx] - l0;
            alpha[0][s] = a0;
        }
    }

    // Pipeline prologue: gather emission inputs for t = 1 (alpha-independent)
    float raw_cur = x[((long long)1 * B_DIM + b) * V_DIM + extIdx];
    float lse_cur = lse[1 * B_DIM + b];

    __syncthreads();

    for (int t = 1; t < T_DIM; ++t) {
        const int cur = t & 1, prv = cur ^ 1;

        // Issue next step's gather + lse load before touching alpha
        float raw_next = 0.0f, lse_next = 0.0f;
        if (t + 1 < T_DIM) {
            const float* rown = x + ((long long)(t + 1) * B_DIM + b) * V_DIM;
            // hardware prefetch of the row two steps ahead (32 lanes x 64 B = 2 KB)
            if ((t + 2 < T_DIM) && (s < 32))
                __builtin_prefetch(rown + (long long)B_DIM * V_DIM + s * 16, 0, 1);
            raw_next = rown[extIdx];
            lse_next = lse[(t + 1) * B_DIM + b];
        }

        if (act) {
            const float em = raw_cur - lse_cur;
            const float a1 = alpha[prv][s];
            const float a2 = (s >= 1) ? alpha[prv][s - 1] : NEGV;
            const float a3 = skip     ? alpha[prv][s - 2] : NEGV;
            alpha[cur][s] = lae3(a1, a2, a3) + em;
        }
        __syncthreads();

        raw_cur = raw_next;
        lse_cur = lse_next;
    }

    if (s == 0) {
        const int last = (T_DIM - 1) & 1;
        const float ll = lae2(alpha[last][S_DIM - 1], alpha[last][S_DIM - 2]);
        nll[b] = -ll / (float)L_DIM;
    }
}

// ---------------------------------------------------------------------------
// Kernel 3: mean over the 64 per-batch losses via V_WMMA_F32_16X16X4_F32.
// A (16x4 f32) holds the 64 values, B = ones => D[m,n] = sum_k A[m,k].
// f32-exact sum through the matrix pipe (single wave, EXEC all ones).
// ---------------------------------------------------------------------------
__global__ __launch_bounds__(32) void ctc_mean_wmma_kernel(const float* __restrict__ nll,
                                                           float* __restrict__ out) {
    const int lane = threadIdx.x;
    // A-matrix 16x4 layout: lanes 0-15 -> M=lane, K={0,1}; lanes 16-31 -> M=lane-16, K={2,3}
    const int m     = lane & 15;
    const int kbase = (lane < 16) ? 0 : 2;
    v2f a;
    a.x = nll[m * 4 + kbase];
    a.y = nll[m * 4 + kbase + 1];
    v2f bmat;
    bmat.x = 1.0f;
    bmat.y = 1.0f;          // all-ones B: layout-independent
    v8f c = {};
    c = __builtin_amdgcn_wmma_f32_16x16x4_f32(false, a, false, bmat,
                                              (short)0, c, false, false);
    // D column 0: lanes 0..15 hold M=0..7 in c[0..7], lanes 16..31 hold M=8..15
    float part = c[0] + c[1] + c[2] + c[3] + c[4] + c[5] + c[6] + c[7];
    float other = __shfl(part, lane ^ 16, 32);
    if (lane == 0) out[0] = (part + other) * (1.0f / (float)B_DIM);
}

extern "C" void kernel_launch(void* const* d_in, const int* in_sizes, int n_in,
                              void* d_out, int out_size, void* d_ws, size_t ws_size,
                              hipStream_t stream) {
    const float* x       = (const float*)d_in[0];   // [T, B, V] f32
    const int*   targets = (const int*)d_in[1];     // [B, L] i32
    float*       out     = (float*)d_out;           // scalar

    float* lse = (float*)d_ws;                      // T*B floats (512 KB)
    float* nll = lse + (size_t)T_DIM * B_DIM;       // B floats

    // Pass 1: 128000 rows, 8 rows per 256-thread block
    ctc_lse_kernel<<<(T_DIM * B_DIM) / 8, 256, 0, stream>>>(x, lse);
    // Pass 2: one workgroup per batch element (pipelined scan)
    ctc_scan_kernel<<<B_DIM, 416, 0, stream>>>(x, targets, lse, nll);
    // Pass 3: WMMA batch mean
    ctc_mean_wmma_kernel<<<1, 32, 0, stream>>>(nll, out);
}